// TotalClusteringLoss_40114994544957
// MI455X (gfx1250) — compile-verified
//
#include <hip/hip_runtime.h>
#include <hip/hip_bf16.h>

typedef __attribute__((ext_vector_type(2))) float v2f;
typedef __attribute__((ext_vector_type(8))) float v8f;

namespace {
constexpr int kD = 128;          // feature dim (reference: 128)
constexpr int kC = 1000;         // NUM_CLASSES
constexpr int kRowPitch = 132;   // LDS row pitch in floats (128 + 4 pad, bank-spread)
// Workspace layout (floats):
//   [0,        128000) sums[C][D]
//   [128000,   129000) counts[C]
//   [129000,   257000) means[C][D]
//   [257000,   258000) dmax bits (uint)
//   [258000,   259000) dmin bits (uint)
//   [259000,   259008) scalars: [0]=W, [1]=B
//   [259008,   259136) global mean g[D]
constexpr int kOffCounts = kC * kD;
constexpr int kOffMeans  = kOffCounts + kC;
constexpr int kOffDmax   = kOffMeans + kC * kD;
constexpr int kOffDmin   = kOffDmax + kC;
constexpr int kOffScal   = kOffDmin + kC;
constexpr int kOffG      = kOffScal + 8;
}

__device__ inline float waveReduceSum(float v) {
  #pragma unroll
  for (int o = 16; o > 0; o >>= 1) v += __shfl_down(v, o, 32);
  return v;  // valid in lane 0
}

// ---------------- init ----------------
__global__ void init_ws(float* __restrict__ sums /* covers sums+counts */,
                        unsigned* __restrict__ dmax, unsigned* __restrict__ dmin,
                        float* __restrict__ scal) {
  const int i = blockIdx.x * blockDim.x + threadIdx.x;
  if (i < kC * kD + kC) sums[i] = 0.0f;
  if (i < kC) { dmax[i] = 0u; dmin[i] = 0x7F7FFFFFu; }  // d2>=0, FLT_MAX bits
  if (i < 8) scal[i] = 0.0f;
}

// ---------------- pass 1: per-class sums & counts (wave per point) ----------------
__global__ __launch_bounds__(256) void pass1_sums(const float* __restrict__ feats,
                                                  const int* __restrict__ labels,
                                                  float* __restrict__ sums,
                                                  float* __restrict__ counts, int N) {
  const int lane  = threadIdx.x & 31;
  const int gwave = (int)((blockIdx.x * blockDim.x + threadIdx.x) >> 5);
  const int nwave = (int)((gridDim.x * blockDim.x) >> 5);
  for (int i = gwave; i < N; i += nwave) {
    const int lab = labels[i];
    const float4 x = *(const float4*)(feats + (size_t)i * kD + lane * 4);
    float* s = sums + (size_t)lab * kD + lane * 4;
    atomicAdd(s + 0, x.x); atomicAdd(s + 1, x.y);
    atomicAdd(s + 2, x.z); atomicAdd(s + 3, x.w);
    if (lane == 0) atomicAdd(&counts[lab], 1.0f);
  }
}

// ---------------- global mean g = (sum_c S_c)/N ----------------
__global__ void global_mean_k(const float* __restrict__ sums, float* __restrict__ g,
                              float invN) {
  const int d = threadIdx.x;  // 128 threads
  float s = 0.0f;
  for (int c = 0; c < kC; ++c) s += sums[(size_t)c * kD + d];
  g[d] = s * invN;
}

// ---------------- per-class: means + B accumulation (wave per class) -------------
__global__ __launch_bounds__(256) void class_stats(const float* __restrict__ sums,
                                                   const float* __restrict__ counts,
                                                   const float* __restrict__ g,
                                                   float* __restrict__ means,
                                                   float* __restrict__ scal) {
  const int lane  = threadIdx.x & 31;
  const int gwave = (int)((blockIdx.x * blockDim.x + threadIdx.x) >> 5);
  const int nwave = (int)((gridDim.x * blockDim.x) >> 5);
  const float4 gm = *(const float4*)(g + lane * 4);
  for (int c = gwave; c < kC; c += nwave) {
    const float n = counts[c];
    const float inv = 1.0f / fmaxf(n, 1.0f);
    const float4 s = *(const float4*)(sums + (size_t)c * kD + lane * 4);
    float4 m; m.x = s.x * inv; m.y = s.y * inv; m.z = s.z * inv; m.w = s.w * inv;
    *(float4*)(means + (size_t)c * kD + lane * 4) = m;
    float loc = (m.x - gm.x) * (m.x - gm.x) + (m.y - gm.y) * (m.y - gm.y) +
                (m.z - gm.z) * (m.z - gm.z) + (m.w - gm.w) * (m.w - gm.w);
    const float tot = waveReduceSum(loc);
    if (lane == 0 && n > 0.0f) atomicAdd(&scal[1], n * tot);  // B
  }
}

// ---------------- pass 2: d2 via WMMA Gram diagonal; W, dmax, dmin --------------
__global__ __launch_bounds__(256) void pass2_wmma(const float* __restrict__ feats,
                                                  const int* __restrict__ labels,
                                                  const float* __restrict__ means,
                                                  unsigned* __restrict__ dmax,
                                                  unsigned* __restrict__ dmin,
                                                  float* __restrict__ Wacc, int N) {
  __shared__ float sdiff[8][16 * kRowPitch];  // 8 waves/block, ~8.25 KB/wave
  const int lane  = threadIdx.x & 31;
  const int wslot = threadIdx.x >> 5;
  float* ld = sdiff[wslot];
  const int gwave = (int)((blockIdx.x * blockDim.x + threadIdx.x) >> 5);
  const int nwave = (int)((gridDim.x * blockDim.x) >> 5);
  const int fullTiles = N >> 4;

  float wtot = 0.0f;

  for (int tile = gwave; tile < fullTiles; tile += nwave) {
    const int pBase = tile << 4;

    // Stage diff = x - means[label] for 16 points into LDS (coalesced loads).
    #pragma unroll
    for (int r = 0; r < 16; ++r) {
      const int lab = labels[pBase + r];
      const float4 x = *(const float4*)(feats + (size_t)(pBase + r) * kD + lane * 4);
      const float4 m = *(const float4*)(means + (size_t)lab * kD + lane * 4);
      const float4 df = make_float4(x.x - m.x, x.y - m.y, x.z - m.z, x.w - m.w);
      *(float4*)(ld + r * kRowPitch + lane * 4) = df;
    }
    asm volatile("s_wait_dscnt 0" ::: "memory");  // CDNA5 split DS counter

    // Gram accumulation: D += A * A^T. A(16x4 f32) and B(4x16 f32) share the
    // same lane/VGPR image when B = A^T, so pass the same registers twice.
    // Lane L holds row (point) L&15; VGPR half split selects K+0/1 vs K+2/3.
    v8f acc = {};
    const float* abase = ld + (lane & 15) * kRowPitch + ((lane < 16) ? 0 : 2);
    #pragma unroll
    for (int k = 0; k < kD; k += 4) {
      const v2f a = *(const v2f*)(abase + k);
      acc = __builtin_amdgcn_wmma_f32_16x16x4_f32(
          /*neg_a=*/false, a, /*neg_b=*/false, a,
          /*c_mod=*/(short)0, acc, /*reuse_a=*/false, /*reuse_b=*/false);
    }

    // Diagonal of C (16x16 f32 layout): (t,t) -> VGPR t, lane t (t<8);
    //                                   VGPR t-8, lane t+16 (t>=8).
    #pragma unroll
    for (int t = 0; t < 16; ++t) {
      const float dv = (t < 8) ? __shfl(acc[t], t, 32)
                               : __shfl(acc[t - 8], t + 16, 32);
      wtot += dv;  // broadcast value; only lane 0's copy is flushed below
      if (lane == 0) {
        const int lab = labels[pBase + t];
        const unsigned bits = __float_as_uint(dv);  // d2 >= 0: uint order == float order
        atomicMax(&dmax[lab], bits);
        atomicMin(&dmin[lab], bits);
      }
    }
  }

  // Scalar tail (N % 16 points), handled by wave 0 only.
  if (gwave == 0) {
    for (int i = fullTiles << 4; i < N; ++i) {
      const int lab = labels[i];
      const float4 x = *(const float4*)(feats + (size_t)i * kD + lane * 4);
      const float4 m = *(const float4*)(means + (size_t)lab * kD + lane * 4);
      const float dx = x.x - m.x, dy = x.y - m.y, dz = x.z - m.z, dw = x.w - m.w;
      const float d2 = waveReduceSum(dx * dx + dy * dy + dz * dz + dw * dw);
      if (lane == 0) {
        wtot += d2;
        const unsigned bits = __float_as_uint(d2);
        atomicMax(&dmax[lab], bits);
        atomicMin(&dmin[lab], bits);
      }
    }
  }

  if (lane == 0) atomicAdd(Wacc, wtot);
}

// ---------------- finalize ----------------
__global__ __launch_bounds__(256) void finalize_k(const float* __restrict__ counts,
                                                  const unsigned* __restrict__ dmax,
                                                  const unsigned* __restrict__ dmin,
                                                  const float* __restrict__ scal,
                                                  float* __restrict__ out) {
  __shared__ float smm[256];
  __shared__ float scnt[256];
  const int tid = threadIdx.x;
  float mm = 0.0f, cnt = 0.0f;
  for (int c = tid; c < kC; c += 256) {
    if (counts[c] > 0.0f) {
      mm += __uint_as_float(dmax[c]) - __uint_as_float(dmin[c]);
      cnt += 1.0f;
    }
  }
  smm[tid] = mm; scnt[tid] = cnt;
  __syncthreads();
  for (int s = 128; s > 0; s >>= 1) {
    if (tid < s) { smm[tid] += smm[tid + s]; scnt[tid] += scnt[tid + s]; }
    __syncthreads();
  }
  if (tid == 0) {
    const float W = scal[0], B = scal[1];
    const float maxmin = smm[0] / fmaxf(scnt[0], 1.0f);
    out[0] = W / (B + 1e-8f) + 0.1f * maxmin;  // WB_WEIGHT=1, MAX_MIN_WEIGHT=0.1
  }
}

extern "C" void kernel_launch(void* const* d_in, const int* in_sizes, int n_in,
                              void* d_out, int out_size, void* d_ws, size_t ws_size,
                              hipStream_t stream) {
  const float* feats = (const float*)d_in[0];
  const int* labels  = (const int*)d_in[1];
  const int N = in_sizes[1];  // labels count

  float* ws = (float*)d_ws;
  float*    sums   = ws;                       // also covers counts in init
  float*    counts = ws + kOffCounts;
  float*    means  = ws + kOffMeans;
  unsigned* dmax   = (unsigned*)(ws + kOffDmax);
  unsigned* dmin   = (unsigned*)(ws + kOffDmin);
  float*    scal   = ws + kOffScal;            // [0]=W, [1]=B
  float*    g      = ws + kOffG;

  init_ws<<<(kC * kD + kC + 255) / 256, 256, 0, stream>>>(sums, dmax, dmin, scal);
  pass1_sums<<<2048, 256, 0, stream>>>(feats, labels, sums, counts, N);
  global_mean_k<<<1, kD, 0, stream>>>(sums, g, 1.0f / (float)N);
  class_stats<<<125, 256, 0, stream>>>(sums, counts, g, means, scal);
  pass2_wmma<<<1024, 256, 0, stream>>>(feats, labels, means, dmax, dmin, &scal[0], N);
  finalize_k<<<1, 256, 0, stream>>>(counts, dmax, dmin, scal, (float*)d_out);
}